// MonarchProjection_24558622998774
// MI455X (gfx1250) — compile-verified
//
#include <hip/hip_runtime.h>

typedef float v2f __attribute__((ext_vector_type(2)));
typedef float v8f __attribute__((ext_vector_type(8)));
typedef unsigned int v4u __attribute__((ext_vector_type(4)));
typedef unsigned int v8u __attribute__((ext_vector_type(8)));

#define SQ     32      // block size s (32x32 blocks, 32 of them)
#define DIN    1024    // d_in = s*s
#define TOK    16      // tokens per workgroup (= WMMA N)
#define PITCH  17      // padded LDS stride for us/vs (gcd(17,64)=1)
#define XPITCH 1028    // TDM-padded x row pitch in floats: 1024 + 4 pads (1 DW / 256 DW)
#define REGION 17408   // floats per LDS region: max(16*1028, 1024*17)

// xs element address (floats): TDM pads 1 DWORD after every 256 DWORDs streamed,
// so sequential element n = t*1024 + d lands at n + n/256 = t*1028 + d + (d>>8).
__device__ __forceinline__ int xsaddr(int t, int d) {
    return t * XPITCH + d + (d >> 8);
}

__global__ __launch_bounds__(256, 1)
void monarch_wmma_kernel(const float* __restrict__ x,
                         const float* __restrict__ L,
                         const float* __restrict__ R,
                         const float* __restrict__ dk,
                         const float* __restrict__ bias,
                         float* __restrict__ out,
                         int n_rows)
{
    extern __shared__ float smem[];
    float* xs = smem;            // [t][d] TDM-padded x tile (region 0)
    float* us = smem + REGION;   // [j*32+n][t], 17-pitch (region 1)
    float* vs = xs;              // stage-2 result, reuses region 0

    const int tid  = threadIdx.x;
    const int wave = tid >> 5;      // 8 waves
    const int lane = tid & 31;
    const int half = lane >> 4;     // 0: lanes 0-15, 1: lanes 16-31
    const int lt   = lane & 15;     // token index within tile for B/C/D frags
    const long long t0 = (long long)blockIdx.x * TOK;

    // ---- TDM: DMA the 16x1024 f32 tile global->LDS with bank-break padding ----
    if (wave == 0) {
        const unsigned long long gaddr = (unsigned long long)(const void*)(x + t0 * DIN);
        const unsigned int lds_base = (unsigned int)(unsigned long long)(void*)xs;

        v4u g0;
        g0[0] = 1u;                                   // count=1 valid, load, no gather
        g0[1] = lds_base;                             // lds_addr (bytes)
        g0[2] = (unsigned int)(gaddr & 0xffffffffu);  // global_addr[31:0]
        g0[3] = (unsigned int)((gaddr >> 32) & 0x1ffffffu) | (2u << 30); // [56:32] | type=2

        v8u g1;
        g1[0] = (2u << 16)        // data_size = 4 bytes
              | (1u << 20)        // pad_enable
              | (7u << 22);       // pad_interval: 256 DWORDs (pad_amount=0 -> 1 DWORD)
        g1[1] = (1024u & 0xffffu) << 16;  // tensor_dim0[15:0] at bits 63:48
        g1[2] = (16u << 16);              // tensor_dim0 hi=0 | tensor_dim1[15:0]=16
        g1[3] = (1024u << 16);            // tensor_dim1 hi=0 | tile_dim0=1024
        g1[4] = 16u;                      // tile_dim1=16 | tile_dim2=0
        g1[5] = 1024u;                    // tensor_dim0_stride[31:0]
        g1[6] = 0u;                       // stride hi | tensor_dim1_stride lo (2D: unused)
        g1[7] = 0u;

        v4u g2 = {0u, 0u, 0u, 0u};        // 2-D tensor: groups 2/3 unused
        v4u g3 = {0u, 0u, 0u, 0u};

        asm volatile("tensor_load_to_lds %0, %1, %2, %3"
                     :: "s"(g0), "s"(g1), "s"(g2), "s"(g3)
                     : "memory");
        __builtin_amdgcn_s_wait_tensorcnt(0);
    }
    __syncthreads();

    // ---- stage 1: per block j:  U_j[n,t] = sum_i L[j,n,i] * x_t[i*32+j] ----
    for (int jj = 0; jj < 4; ++jj) {
        const int j = wave + jj * 8;            // uniform per wave
        const float* Lj = L + j * (SQ * SQ);    // L[j][n][i]
        v8f acc0 = {}; v8f acc1 = {};
        #pragma unroll
        for (int k = 0; k < 8; ++k) {
            const int i0 = 4 * k + 2 * half;    // A/B K-split across lane halves
            v2f b;
            b.x = xs[xsaddr(lt, (i0    ) * SQ + j)];
            b.y = xs[xsaddr(lt, (i0 + 1) * SQ + j)];
            v2f a0 = *(const v2f*)(Lj + (     lt) * SQ + i0);   // M-tile 0: n=lt
            v2f a1 = *(const v2f*)(Lj + (16 + lt) * SQ + i0);   // M-tile 1: n=16+lt
            acc0 = __builtin_amdgcn_wmma_f32_16x16x4_f32(false, a0, false, b,
                                                         (short)0, acc0, false, false);
            acc1 = __builtin_amdgcn_wmma_f32_16x16x4_f32(false, a1, false, b,
                                                         (short)0, acc1, false, false);
        }
        #pragma unroll
        for (int v = 0; v < 8; ++v) {           // C/D layout: M = v + 8*half (+16 tile 1)
            us[(j * SQ + (     8 * half + v)) * PITCH + lt] = acc0[v];
            us[(j * SQ + (16 + 8 * half + v)) * PITCH + lt] = acc1[v];
        }
    }
    __syncthreads();

    // ---- stage 2: per block n:  V_n[p,t] = sum_j R[n,p,j] * U_j[n,t] ----
    for (int nn = 0; nn < 4; ++nn) {
        const int n = wave + nn * 8;
        const float* Rn = R + n * (SQ * SQ);    // R[n][p][j]
        v8f acc0 = {}; v8f acc1 = {};
        #pragma unroll
        for (int k = 0; k < 8; ++k) {
            const int j0 = 4 * k + 2 * half;
            v2f b;
            b.x = us[((j0    ) * SQ + n) * PITCH + lt];
            b.y = us[((j0 + 1) * SQ + n) * PITCH + lt];
            v2f a0 = *(const v2f*)(Rn + (     lt) * SQ + j0);
            v2f a1 = *(const v2f*)(Rn + (16 + lt) * SQ + j0);
            acc0 = __builtin_amdgcn_wmma_f32_16x16x4_f32(false, a0, false, b,
                                                         (short)0, acc0, false, false);
            acc1 = __builtin_amdgcn_wmma_f32_16x16x4_f32(false, a1, false, b,
                                                         (short)0, acc1, false, false);
        }
        #pragma unroll
        for (int v = 0; v < 8; ++v) {           // output element d = p*32 + n
            int p0 =      8 * half + v;
            int p1 = 16 + 8 * half + v;
            vs[(p0 * SQ + n) * PITCH + lt] = acc0[v];
            vs[(p1 * SQ + n) * PITCH + lt] = acc1[v];
        }
    }
    __syncthreads();

    // ---- gating epilogue: out = d_kernel[row] * y + b, fully coalesced ----
    {
        const long long base_out = t0 * DIN;
        const int row0 = (int)(t0 % n_rows);    // tile never crosses batch boundary
        #pragma unroll 4
        for (int q = 0; q < (TOK * DIN) / 256; ++q) {
            int e = q * 256 + tid;
            int t = e >> 10;
            int d = e & (DIN - 1);
            float val = vs[d * PITCH + t];      // stride-17 -> conflict-free gather
            out[base_out + e] = dk[(long long)(row0 + t) * DIN + d] * val + bias[d];
        }
    }
}

extern "C" void kernel_launch(void* const* d_in, const int* in_sizes, int n_in,
                              void* d_out, int out_size, void* d_ws, size_t ws_size,
                              hipStream_t stream) {
    const float* x    = (const float*)d_in[0];
    const float* L    = (const float*)d_in[1];
    const float* R    = (const float*)d_in[2];
    const float* dk   = (const float*)d_in[3];
    const float* bias = (const float*)d_in[4];
    float* out = (float*)d_out;

    const int tokens = in_sizes[0] / DIN;   // b_sz * n = 16384
    const int n_rows = in_sizes[3] / DIN;   // 4096
    const int blocks = tokens / TOK;        // 1024 workgroups
    const size_t shmem = (size_t)2 * REGION * sizeof(float);  // ~136 KB

    monarch_wmma_kernel<<<blocks, 256, shmem, stream>>>(x, L, R, dk, bias, out, n_rows);
}